// CompGCN_ConvE_10445360463971
// MI455X (gfx1250) — compile-verified
//
#include <hip/hip_runtime.h>
#include <math.h>

// ---------------------------------------------------------------------------
// CompGCN (6 layers) + ConvE decoder for MI455X (gfx1250, wave32, WMMA).
// All GEMMs run on V_WMMA_F32_16X16X4_F32 (exact fp32 semantics).
// GEMM: one wave computes a 64x16 tile (4 M-subtiles share each B fragment).
// Decoder: conv fused with the 18816-K FC; fc_w is pre-transposed+padded so
// the WMMA k-loops are branch-free and B loads are lane-coalesced.
// ---------------------------------------------------------------------------

static constexpr int D0   = 200;   // feature dim
static constexpr int DP   = 208;   // padded feature dim (13 tiles of 16)
static constexpr int NTN  = 13;    // N tiles in padded dim
static constexpr int NE   = 8192;  // edges
static constexpr int NV   = 5000;  // nodes
static constexpr int NVP  = 5056;  // padded nodes (multiple of 64)
static constexpr int NLAY = 6;
static constexpr int NF   = 96;    // conv channels
static constexpr int CONVK= 49;    // 7x7
static constexpr int P196 = 196;   // 14*14 per-channel conv outputs
static constexpr int FLAT = 18816; // 96*196
static constexpr int NCLS = 13;
#define INV_SQRT_1PE 0.99999500f   // 1/sqrt(1+1e-5)

typedef __attribute__((ext_vector_type(2))) float v2f;
typedef __attribute__((ext_vector_type(8))) float v8f;

static __device__ __forceinline__ v8f wmma4(v2f a, v2f b, v8f c) {
  // D = A(16x4,f32) * B(4x16,f32) + C(16x16,f32)
  return __builtin_amdgcn_wmma_f32_16x16x4_f32(false, a, false, b, (short)0, c,
                                               false, false);
}

// ------------------------------- elementwise --------------------------------

__global__ void k_zero(float* __restrict__ p, int n) {
  int i = blockIdx.x * blockDim.x + threadIdx.x;
  if (i < n) p[i] = 0.0f;
}

// pad [rows,200] -> [prows,208] with zero fill
__global__ void k_pad(const float* __restrict__ src, float* __restrict__ dst,
                      int rows, int prows) {
  int i = blockIdx.x * blockDim.x + threadIdx.x;
  int total = prows * DP;
  if (i >= total) return;
  int r = i / DP, c = i % DP;
  dst[i] = (r < rows && c < D0) ? src[r * D0 + c] : 0.0f;
}

// pack Win/Wout/Wloop/Wrel of layer l into WP[4][208][208], zero-padded
__global__ void k_pad_w(const float* __restrict__ Win, const float* __restrict__ Wout,
                        const float* __restrict__ Wloop, const float* __restrict__ Wrel,
                        int l, float* __restrict__ WP) {
  int i = blockIdx.x * blockDim.x + threadIdx.x;
  int total = 4 * DP * DP;
  if (i >= total) return;
  int w = i / (DP * DP);
  int rem = i % (DP * DP);
  int r = rem / DP, c = rem % DP;
  const float* src = (w == 0 ? Win : w == 1 ? Wout : w == 2 ? Wloop : Wrel)
                     + (size_t)l * D0 * D0;
  WP[i] = (r < D0 && c < D0) ? src[r * D0 + c] : 0.0f;
}

// fcwT[k][n] = fc_w[n][k] (n<200, else 0) -- one-time transpose+pad
__global__ void k_tr_fcw(const float* __restrict__ fcw, float* __restrict__ fcwT) {
  int i = blockIdx.x * blockDim.x + threadIdx.x;
  if (i >= FLAT * DP) return;
  int k = i / DP, n = i % DP;
  fcwT[i] = (n < D0) ? fcw[(size_t)n * FLAT + k] : 0.0f;
}

__global__ void k_deg(const int* __restrict__ ei0, const int* __restrict__ iei0,
                      float* __restrict__ degf, float* __restrict__ degi) {
  int e = blockIdx.x * blockDim.x + threadIdx.x;
  if (e < NE)            atomicAdd(&degf[ei0[e]], 1.0f);
  else if (e < 2 * NE)   atomicAdd(&degi[iei0[e - NE]], 1.0f);
}

__global__ void k_norm(const int* __restrict__ ei0, const int* __restrict__ ei1,
                       const int* __restrict__ iei0, const int* __restrict__ iei1,
                       const float* __restrict__ degf, const float* __restrict__ degi,
                       float* __restrict__ nf, float* __restrict__ ni) {
  int e = blockIdx.x * blockDim.x + threadIdx.x;
  if (e >= NE) return;
  float df0 = degf[ei0[e]],  df1 = degf[ei1[e]];
  float di0 = degi[iei0[e]], di1 = degi[iei1[e]];
  float a = df0 > 0.f ? rsqrtf(fmaxf(df0, 1.f)) : 0.f;
  float b = df1 > 0.f ? rsqrtf(fmaxf(df1, 1.f)) : 0.f;
  float c = di0 > 0.f ? rsqrtf(fmaxf(di0, 1.f)) : 0.f;
  float d = di1 > 0.f ? rsqrtf(fmaxf(di1, 1.f)) : 0.f;
  nf[e] = a * b;
  ni[e] = c * d;
}

// ain = x[ei0] - r ; aout = x[iei0] - ir  (over padded [NE,208])
__global__ void k_gather(const float* __restrict__ xP, const float* __restrict__ r,
                         const float* __restrict__ ir,
                         const int* __restrict__ ei0, const int* __restrict__ iei0,
                         float* __restrict__ ain, float* __restrict__ aout) {
  int i = blockIdx.x * blockDim.x + threadIdx.x;
  if (i >= NE * DP) return;
  int e = i / DP, c = i % DP;
  ain[i]  = xP[(size_t)ei0[e] * DP + c] - r[i];
  aout[i] = xP[(size_t)iei0[e] * DP + c] - ir[i];
}

// agg[dstidx[e]] += src[e]  (float atomics)
__global__ void k_scatter(const float* __restrict__ src, const int* __restrict__ dstidx,
                          float* __restrict__ agg) {
  int i = blockIdx.x * blockDim.x + threadIdx.x;
  if (i >= NE * DP) return;
  int e = i / DP, c = i % DP;
  atomicAdd(&agg[(size_t)dstidx[e] * DP + c], src[i]);
}

// x = tanh((agg + x@Wloop)/3 + b_gcn[l])
__global__ void k_update(float* __restrict__ xP, const float* __restrict__ agg,
                         const float* __restrict__ xl, const float* __restrict__ b_gcn,
                         int l) {
  int i = blockIdx.x * blockDim.x + threadIdx.x;
  if (i >= NV * D0) return;
  int r = i / D0, c = i % D0;
  size_t o = (size_t)r * DP + c;
  xP[o] = tanhf((agg[o] + xl[o]) * (1.0f / 3.0f) + b_gcn[l * D0 + c]);
}

// ------------------------------ WMMA GEMM -----------------------------------
// C[M,208] = A[M,208] @ B[208,208] (row-major, stride 208). One wave computes
// a 64x16 tile: 4 M-subtiles share each B fragment. Optional per-row scale
// (degree normalization) is a template parameter -> branch-free epilogue.
// grid.x = (M/64) * 13, block = 32.
template <bool HAS_SCALE>
__global__ __launch_bounds__(32) void k_gemm(const float* __restrict__ A,
                                             const float* __restrict__ B,
                                             float* __restrict__ C,
                                             const float* __restrict__ rowscale) {
  int tile = blockIdx.x;
  int tm = tile / NTN, tn = tile % NTN;
  int m0 = tm * 64, n0 = tn * 16;
  int lane = threadIdx.x;
  int half = lane >> 4;        // K-group: lanes 0-15 -> K {k,k+1}; 16-31 -> {k+2,k+3}
  int l16  = lane & 15;

  const float* A0 = A + (size_t)(m0 + l16) * DP;
  const float* Bc = B + n0 + l16;
  v8f acc0 = {}, acc1 = {}, acc2 = {}, acc3 = {};
#pragma unroll 2
  for (int k = 0; k < DP; k += 4) {
    int kh = k + 2 * half;
    v2f b;
    b.x = Bc[(size_t)kh * DP];                                   // B[k][n]
    b.y = Bc[(size_t)(kh + 1) * DP];
    v2f a0 = *reinterpret_cast<const v2f*>(A0 + kh);             // A[m][k..k+1]
    v2f a1 = *reinterpret_cast<const v2f*>(A0 + 16 * DP + kh);
    v2f a2 = *reinterpret_cast<const v2f*>(A0 + 32 * DP + kh);
    v2f a3 = *reinterpret_cast<const v2f*>(A0 + 48 * DP + kh);
    acc0 = wmma4(a0, b, acc0);
    acc1 = wmma4(a1, b, acc1);
    acc2 = wmma4(a2, b, acc2);
    acc3 = wmma4(a3, b, acc3);
  }
#pragma unroll
  for (int s = 0; s < 4; ++s) {
    v8f acc = s == 0 ? acc0 : s == 1 ? acc1 : s == 2 ? acc2 : acc3;
    int mb = m0 + 16 * s + 8 * half;
#pragma unroll
    for (int v = 0; v < 8; ++v) {
      int m = mb + v;                                            // C/D row layout
      float val = acc[v];
      if (HAS_SCALE) val *= rowscale[m];
      C[(size_t)m * DP + n0 + l16] = val;
    }
  }
}

// --------------------- fused ConvE decoder (conv+FC+fc1) --------------------
// One block = 16 edges, 4 waves. Per channel c: conv (16x196) into LDS, then
// consumed as WMMA-A against fcwT (K-chunk = 196 = 49 exact k-steps).
// Wave w owns N-tiles {w, w+4, w+8}; wave 0 additionally owns tile 12 (valid
// because fcwT is padded to 208 columns). The per-wave tile count is decided
// by a wave-uniform branch OUTSIDE the k-loop so each k-loop body is
// straight-line code (in-place WMMA accumulation, no live-range splits).
__global__ __launch_bounds__(128) void k_convfc(
    const float* __restrict__ xP, const float* __restrict__ rP,
    const int* __restrict__ ei0, const float* __restrict__ fcwT,
    const float* __restrict__ conv_w, const float* __restrict__ conv_b,
    const float* __restrict__ bn1_g, const float* __restrict__ bn1_b,
    const float* __restrict__ fc_b,
    const float* __restrict__ bn2_g, const float* __restrict__ bn2_b,
    const float* __restrict__ fc1_w, const float* __restrict__ fc1_b,
    const float* __restrict__ bn0_g, const float* __restrict__ bn0_b,
    float* __restrict__ out) {
  __shared__ float img[16][400];    // 20x20 interleaved (sub,r) after bn0
  __shared__ float hbuf[16][P196];  // per-channel conv output (post bn1+relu)
  __shared__ float h2s[16][D0];     // FC output after bn2+relu
  __shared__ float cw[CONVK];

  int m0   = blockIdx.x * 16;
  int tid  = threadIdx.x;
  int wave = tid >> 5, lane = tid & 31;
  int half = lane >> 4, l16 = lane & 15;
  int nb   = wave * 16;             // first N-tile column offset for this wave

  float b0s = bn0_g[0] * INV_SQRT_1PE;
  float b0b = bn0_b[0];

  // build image tiles: flat k even -> sub[k/2], odd -> r[k/2]
  for (int i = tid; i < 16 * 400; i += 128) {
    int t = i / 400, k = i % 400;
    int src = ei0[m0 + t];
    float v = (k & 1) ? rP[(size_t)(m0 + t) * DP + (k >> 1)]
                      : xP[(size_t)src * DP + (k >> 1)];
    img[t][k] = v * b0s + b0b;
  }
  __syncthreads();

  v8f acc0 = {}, acc1 = {}, acc2 = {}, acc3 = {};

  for (int c = 0; c < NF; ++c) {
    if (tid < CONVK) cw[tid] = conv_w[c * CONVK + tid];
    __syncthreads();

    float sc1 = bn1_g[c] * INV_SQRT_1PE;
    float bb1 = bn1_b[c];
    float cb  = conv_b[c];
    for (int i = tid; i < 16 * P196; i += 128) {
      int t = i / P196, p = i % P196;
      int oy = p / 14, ox = p % 14;
      float s = cb;
#pragma unroll
      for (int a = 0; a < 7; ++a)
#pragma unroll
        for (int b = 0; b < 7; ++b)
          s += img[t][(oy + a) * 20 + ox + b] * cw[a * 7 + b];
      float v = s * sc1 + bb1;
      hbuf[t][p] = v > 0.f ? v : 0.f;
    }
    __syncthreads();

    const int kg0 = c * P196;
    const float* hrow = &hbuf[l16][0];
    if (wave == 0) {
      // wave-uniform: 4 tiles {0,4,8,12}; straight-line body, EXEC all-ones
      for (int kk = 0; kk < P196; kk += 4) {
        v2f a = *reinterpret_cast<const v2f*>(hrow + kk + 2 * half);
        const float* Bk = fcwT + (size_t)(kg0 + kk + 2 * half) * DP + l16;
        v2f b0, b1, b2, b3;
        b0.x = Bk[0];        b0.y = Bk[DP + 0];
        b1.x = Bk[64];       b1.y = Bk[DP + 64];
        b2.x = Bk[128];      b2.y = Bk[DP + 128];
        b3.x = Bk[192];      b3.y = Bk[DP + 192];
        acc0 = wmma4(a, b0, acc0);
        acc1 = wmma4(a, b1, acc1);
        acc2 = wmma4(a, b2, acc2);
        acc3 = wmma4(a, b3, acc3);
      }
    } else {
      // waves 1..3: tiles {w, w+4, w+8}
      for (int kk = 0; kk < P196; kk += 4) {
        v2f a = *reinterpret_cast<const v2f*>(hrow + kk + 2 * half);
        const float* Bk = fcwT + (size_t)(kg0 + kk + 2 * half) * DP + l16 + nb;
        v2f b0, b1, b2;
        b0.x = Bk[0];        b0.y = Bk[DP + 0];
        b1.x = Bk[64];       b1.y = Bk[DP + 64];
        b2.x = Bk[128];      b2.y = Bk[DP + 128];
        acc0 = wmma4(a, b0, acc0);
        acc1 = wmma4(a, b1, acc1);
        acc2 = wmma4(a, b2, acc2);
      }
    }
    __syncthreads();
  }

  // epilogue: h2 = relu((acc + fc_b) * bn2_scale + bn2_b)
  auto store_tile = [&](v8f acc, int tn) {
    int n = tn * 16 + l16;
    if (n < D0) {
      float sc2 = bn2_g[n] * INV_SQRT_1PE;
      float bb2 = bn2_b[n];
      float fb  = fc_b[n];
#pragma unroll
      for (int v = 0; v < 8; ++v) {
        float val = (acc[v] + fb) * sc2 + bb2;
        h2s[v + 8 * half][n] = val > 0.f ? val : 0.f;
      }
    }
  };
  store_tile(acc0, wave);
  store_tile(acc1, wave + 4);
  store_tile(acc2, wave + 8);
  if (wave == 0) store_tile(acc3, 12);
  __syncthreads();

  // fc1 head: 16 edges x 13 classes
  for (int i = tid; i < 16 * NCLS; i += 128) {
    int t = i / NCLS, cls = i % NCLS;
    float s = fc1_b[cls];
    for (int c2 = 0; c2 < D0; ++c2) s += h2s[t][c2] * fc1_w[cls * D0 + c2];
    out[(size_t)(m0 + t) * NCLS + cls] = s;
  }
}

// ------------------------------- host side ----------------------------------

extern "C" void kernel_launch(void* const* d_in, const int* in_sizes, int n_in,
                              void* d_out, int out_size, void* d_ws, size_t ws_size,
                              hipStream_t stream) {
  const float* node  = (const float*)d_in[0];
  const float* ef    = (const float*)d_in[1];
  const float* ief   = (const float*)d_in[2];
  const float* Win   = (const float*)d_in[3];
  const float* Wout  = (const float*)d_in[4];
  const float* Wloop = (const float*)d_in[5];
  const float* Wrel  = (const float*)d_in[6];
  const float* bgcn  = (const float*)d_in[7];
  const float* bn0g  = (const float*)d_in[8];
  const float* bn0b  = (const float*)d_in[9];
  const float* convw = (const float*)d_in[10];
  const float* convb = (const float*)d_in[11];
  const float* bn1g  = (const float*)d_in[12];
  const float* bn1b  = (const float*)d_in[13];
  const float* fcw   = (const float*)d_in[14];
  const float* fcb   = (const float*)d_in[15];
  const float* bn2g  = (const float*)d_in[16];
  const float* bn2b  = (const float*)d_in[17];
  const float* fc1w  = (const float*)d_in[18];
  const float* fc1b  = (const float*)d_in[19];
  const int*   ei    = (const int*)d_in[20];
  const int*   iei   = (const int*)d_in[21];
  const int *ei0 = ei, *ei1 = ei + NE;
  const int *iei0 = iei, *iei1 = iei + NE;

  float* ws = (float*)d_ws;
  size_t o = 0;
  float* xP   = ws + o; o += (size_t)NVP * DP;
  float* xlP  = ws + o; o += (size_t)NVP * DP;
  float* aggP = ws + o; o += (size_t)NVP * DP;
  float* r0   = ws + o; o += (size_t)NE * DP;
  float* r1   = ws + o; o += (size_t)NE * DP;
  float* ir0  = ws + o; o += (size_t)NE * DP;
  float* ir1  = ws + o; o += (size_t)NE * DP;
  float* ain  = ws + o; o += (size_t)NE * DP;
  float* aout = ws + o; o += (size_t)NE * DP;
  float* minb = ws + o; o += (size_t)NE * DP;
  float* moutb= ws + o; o += (size_t)NE * DP;
  float* WP   = ws + o; o += (size_t)4 * DP * DP;
  float* fcwT = ws + o; o += (size_t)FLAT * DP;
  float* degf = ws + o; o += NVP;
  float* degi = ws + o; o += NVP;
  float* nfv  = ws + o; o += NE;
  float* niv  = ws + o; o += NE;

  const int TB = 256;
  auto cdiv = [](int a, int b) { return (a + b - 1) / b; };

  // padded inputs + transposed fc_w (independent of the GCN loop)
  k_pad<<<cdiv(NVP * DP, TB), TB, 0, stream>>>(node, xP, NV, NVP);
  k_pad<<<cdiv(NE * DP, TB), TB, 0, stream>>>(ef, r0, NE, NE);
  k_pad<<<cdiv(NE * DP, TB), TB, 0, stream>>>(ief, ir0, NE, NE);
  k_tr_fcw<<<cdiv(FLAT * DP, TB), TB, 0, stream>>>(fcw, fcwT);

  // degree norms
  k_zero<<<cdiv(2 * NVP, TB), TB, 0, stream>>>(degf, 2 * NVP);  // degf+degi contiguous
  k_deg<<<cdiv(2 * NE, TB), TB, 0, stream>>>(ei0, iei0, degf, degi);
  k_norm<<<cdiv(NE, TB), TB, 0, stream>>>(ei0, ei1, iei0, iei1, degf, degi, nfv, niv);

  float *rc = r0, *rn = r1, *irc = ir0, *irn = ir1;
  const int GE = (NE / 64) * NTN;   // edge-GEMM tiles (64-row blocks)
  const int GV = (NVP / 64) * NTN;  // node-GEMM tiles

  for (int l = 0; l < NLAY; ++l) {
    k_pad_w<<<cdiv(4 * DP * DP, TB), TB, 0, stream>>>(Win, Wout, Wloop, Wrel, l, WP);
    k_gather<<<cdiv(NE * DP, TB), TB, 0, stream>>>(xP, rc, irc, ei0, iei0, ain, aout);
    k_gemm<true><<<GE, 32, 0, stream>>>(ain,  WP + 0 * DP * DP, minb,  nfv);
    k_gemm<true><<<GE, 32, 0, stream>>>(aout, WP + 1 * DP * DP, moutb, niv);
    k_gemm<false><<<GV, 32, 0, stream>>>(xP,   WP + 2 * DP * DP, xlP,   nullptr);
    k_gemm<false><<<GE, 32, 0, stream>>>(rc,   WP + 3 * DP * DP, rn,    nullptr);
    k_gemm<false><<<GE, 32, 0, stream>>>(irc,  WP + 3 * DP * DP, irn,   nullptr);
    k_zero<<<cdiv(NVP * DP, TB), TB, 0, stream>>>(aggP, NVP * DP);
    k_scatter<<<cdiv(NE * DP, TB), TB, 0, stream>>>(minb,  ei1,  aggP);
    k_scatter<<<cdiv(NE * DP, TB), TB, 0, stream>>>(moutb, iei1, aggP);
    k_update<<<cdiv(NV * D0, TB), TB, 0, stream>>>(xP, aggP, xlP, bgcn, l);
    float* t;
    t = rc; rc = rn; rn = t;
    t = irc; irc = irn; irn = t;
  }

  // fused ConvE decoder
  k_convfc<<<NE / 16, 128, 0, stream>>>(xP, rc, ei0, fcwT, convw, convb,
                                        bn1g, bn1b, fcb, bn2g, bn2b,
                                        fc1w, fc1b, bn0g, bn0b, (float*)d_out);
}